// NEW3DCNN_89421219102889
// MI455X (gfx1250) — compile-verified
//
#include <hip/hip_runtime.h>
#include <hip/hip_bf16.h>
#include <math.h>

typedef __attribute__((ext_vector_type(2))) float v2f;
typedef __attribute__((ext_vector_type(8))) float v8f;

// ---------------------------------------------------------------------------
// Utility: zero a small fp32 buffer (per-layer BN statistics accumulators)
// ---------------------------------------------------------------------------
__global__ void zero_f32(float* __restrict__ p, int n) {
    int i = blockIdx.x * blockDim.x + threadIdx.x;
    if (i < n) p[i] = 0.f;
}

// ---------------------------------------------------------------------------
// Pack conv weights (Cout,Cin,3,3,3) into a zero-padded, K-major layout
// wTr[(tap*CinP + cin) * CoutP + m]  so that WMMA A-fragment loads are fully
// coalesced (16 consecutive lanes read 16 consecutive floats) and padded
// rows/cols are exactly zero (no masking needed on the A side).
// ---------------------------------------------------------------------------
__global__ void pack_weights(const float* __restrict__ Wt, float* __restrict__ wTr,
                             int Cin, int Cout, int CinP, int CoutP)
{
    int idx = blockIdx.x * blockDim.x + threadIdx.x;
    int tot = 27 * CinP * CoutP;
    if (idx >= tot) return;
    int m = idx % CoutP;
    int k = idx / CoutP;
    int cin = k % CinP;
    int tap = k / CinP;
    float v = 0.f;
    if (cin < Cin && m < Cout) v = Wt[((long)m * Cin + cin) * 27 + tap];
    wTr[idx] = v;
}

// ---------------------------------------------------------------------------
// Implicit-GEMM 3D conv (k=3, pad=1, stride=1) in full fp32 via
// V_WMMA_F32_16X16X4_F32, fully specialized per layer:
//   CIN  = input channels, COUT = output channels, LD = log2(spatial dim)
// GEMM: M=COUT (padded to 16), K reordered as (tap, cin), N = 2*D*H*W.
// Block = 128 threads = 4 waves; each wave owns one 16x16 C tile; waves tile
// N.  Loads are branch-free (clamped address + value select).  Fuses bias add
// and per-channel sum/sumsq accumulation for training-mode BatchNorm.
// ---------------------------------------------------------------------------
template <int CIN, int COUT, int LD>
__global__ __launch_bounds__(128)
void conv3d_bn_wmma(const float* __restrict__ X, const float* __restrict__ wTr,
                    const float* __restrict__ bias, float* __restrict__ Y,
                    float* __restrict__ stats)   // [2*COUT]: sums, then sumsq
{
    constexpr int W    = 1 << LD;
    constexpr int HW   = 1 << (2 * LD);
    constexpr int DHW  = 1 << (3 * LD);
    constexpr int CINP = (CIN + 3) & ~3;
    constexpr int COUTP = (COUT + 15) & ~15;

    __shared__ float s_sum[16];
    __shared__ float s_sq[16];
    const int tid  = threadIdx.x;
    const int wave = tid >> 5;
    const int lane = tid & 31;
    if (tid < 16) { s_sum[tid] = 0.f; s_sq[tid] = 0.f; }
    __syncthreads();

    const int m_base = blockIdx.y * 16;
    const int n_base = blockIdx.x * 64 + wave * 16;
    const int ln = lane & 15;   // position within 16-lane half
    const int kg = lane >> 4;   // K-group (ISA A/B layout: 0 -> K0/K1, 1 -> K2/K3)

    // flat output index -> (batch, z, y, x) with pure shifts/masks
    const int n     = n_base + ln;
    const int x     = n & (W - 1);
    const int yy    = (n >> LD) & (W - 1);
    const int z     = (n >> (2 * LD)) & (W - 1);
    const int batch = n >> (3 * LD);

    const float* Xb   = X + (long)batch * (CIN * DHW);
    const float* Arow = wTr + m_base + ln;   // coalesced over ln

    v8f acc = {};
    #pragma unroll 1
    for (int kd = 0; kd < 3; ++kd) {
        const int  zz = z + kd - 1;
        const bool vz = (unsigned)zz < (unsigned)W;
        #pragma unroll 1
        for (int kh = 0; kh < 3; ++kh) {
            const int  y2 = yy + kh - 1;
            const bool vy = vz && ((unsigned)y2 < (unsigned)W);
            const int  rowOff = zz * HW + y2 * W;
            #pragma unroll
            for (int kw = 0; kw < 3; ++kw) {
                const int  x2  = x + kw - 1;
                const bool inb = vy && ((unsigned)x2 < (unsigned)W);
                const int  sOff = inb ? (rowOff + x2) : 0;   // clamped, always valid
                const int  tap  = (kd * 3 + kh) * 3 + kw;
                const float* Ak = Arow + (tap * CINP + kg * 2) * COUTP;
                #pragma unroll
                for (int c0 = 0; c0 < CINP; c0 += 4) {
                    v2f a, b;
                    #pragma unroll
                    for (int j = 0; j < 2; ++j) {
                        a[j] = Ak[(c0 + j) * COUTP];   // zero-padded, no mask
                        const int cin = c0 + 2 * kg + j;
                        int cinc = cin;
                        if constexpr (CIN % 4 != 0)
                            cinc = (cin < CIN) ? cin : 0;          // clamp address
                        float bv = Xb[(long)cinc * DHW + sOff];
                        if constexpr (CIN % 4 != 0)
                            bv = (cin < CIN) ? bv : 0.f;           // zero value
                        b[j] = inb ? bv : 0.f;
                    }
                    acc = __builtin_amdgcn_wmma_f32_16x16x4_f32(
                              /*neg_a=*/false, a, /*neg_b=*/false, b,
                              /*c_mod=*/(short)0, acc,
                              /*reuse_a=*/false, /*reuse_b=*/false);
                }
            }
        }
    }

    // Epilogue: bias, store pre-BN activation, per-channel stats reduction
    const long outB = (long)batch * (COUT * DHW) + (long)z * HW + (long)yy * W + x;
    #pragma unroll
    for (int r = 0; r < 8; ++r) {
        const int mc = r + 8 * kg;          // local channel (C/D VGPR layout)
        const int m  = m_base + mc;
        float v = acc[r];
        const bool valid = (COUT % 16 == 0) ? true : (m < COUT);
        if (valid) {
            v += bias[m];
            Y[outB + (long)m * DHW] = v;
        } else {
            v = 0.f;
        }
        float s = v, s2 = v * v;
        #pragma unroll
        for (int off = 1; off < 16; off <<= 1) {   // reduce within 16-lane half
            s  += __shfl_xor(s,  off, 32);
            s2 += __shfl_xor(s2, off, 32);
        }
        if (ln == 0 && valid) {
            atomicAdd(&s_sum[mc], s);
            atomicAdd(&s_sq[mc],  s2);
        }
    }
    __syncthreads();
    if (tid < 16) {
        const int m = m_base + tid;
        if (m < COUT) {
            atomicAdd(&stats[m],        s_sum[tid]);
            atomicAdd(&stats[COUT + m], s_sq[tid]);
        }
    }
}

// ---------------------------------------------------------------------------
// BN stats -> per-channel scale/shift: scale=g*rsqrt(var+eps),
// shift=beta-mean*scale
// ---------------------------------------------------------------------------
__global__ void bn_finalize(const float* __restrict__ stats,
                            const float* __restrict__ g,
                            const float* __restrict__ be,
                            float* __restrict__ scsh, int C, float invCnt)
{
    int c = blockIdx.x * blockDim.x + threadIdx.x;
    if (c >= C) return;
    float mean = stats[c] * invCnt;
    float var  = stats[C + c] * invCnt - mean * mean;
    float sc   = g[c] * rsqrtf(var + 1e-5f);
    scsh[c]     = sc;
    scsh[C + c] = be[c] - mean * sc;
}

// ---------------------------------------------------------------------------
// Fused BN(affine) + ReLU + 2x2x2 max-pool.  Do is the OUTPUT dim.
// ---------------------------------------------------------------------------
__global__ void bn_relu_pool_k(const float* __restrict__ Y,
                               const float* __restrict__ scsh,
                               float* __restrict__ O, int C, int Do)
{
    const int Ho = Do, Wo = Do;
    const long tot = 2L * C * Do * Ho * Wo;
    long idx = (long)blockIdx.x * blockDim.x + threadIdx.x;
    if (idx >= tot) return;
    int x = idx % Wo; long t = idx / Wo;
    int y = t % Ho;  t /= Ho;
    int z = t % Do;  t /= Do;
    int c = t % C;   int nb = (int)(t / C);
    const int Hi = 2 * Ho, Wi = 2 * Wo, Di = 2 * Do;
    const float sc = scsh[c], sh = scsh[C + c];
    const float* base = Y + (((long)nb * C + c) * Di + 2 * z) * Hi * Wi
                          + (long)(2 * y) * Wi + 2 * x;
    float m = -3.4e38f;
    #pragma unroll
    for (int dz = 0; dz < 2; ++dz)
        #pragma unroll
        for (int dy = 0; dy < 2; ++dy)
            #pragma unroll
            for (int dx = 0; dx < 2; ++dx) {
                float v = base[(long)dz * Hi * Wi + dy * Wi + dx] * sc + sh;
                m = fmaxf(m, v);
            }
    O[idx] = fmaxf(m, 0.f);   // ReLU commutes with max (monotone)
}

// ---------------------------------------------------------------------------
// Block 5 tail: BN + ReLU + global average over 8^3=512 voxels -> feats[2*C]
// ---------------------------------------------------------------------------
__global__ __launch_bounds__(512)
void bn_relu_avg_k(const float* __restrict__ Y, const float* __restrict__ scsh,
                   float* __restrict__ feats, int C)
{
    __shared__ float red[512];
    const int nc = blockIdx.x;
    const int c  = nc % C;
    const float sc = scsh[c], sh = scsh[C + c];
    float v = fmaxf(Y[(long)nc * 512 + threadIdx.x] * sc + sh, 0.f);
    red[threadIdx.x] = v;
    __syncthreads();
    for (int s = 256; s > 0; s >>= 1) {
        if ((int)threadIdx.x < s) red[threadIdx.x] += red[threadIdx.x + s];
        __syncthreads();
    }
    if (threadIdx.x == 0) feats[nc] = red[0] * (1.f / 512.f);
}

// ---------------------------------------------------------------------------
// MLP head: feats[2,128] -> fc1(512)+ReLU -> fc2(9) -> sigmoid interval map.
// Writes xp (2x9) at out[3..20].
// ---------------------------------------------------------------------------
__global__ __launch_bounds__(512)
void head_k(const float* __restrict__ feats,
            const float* __restrict__ w1, const float* __restrict__ b1,
            const float* __restrict__ w2, const float* __restrict__ b2,
            float* __restrict__ out)
{
    __shared__ float f[2 * 128];
    __shared__ float h[2 * 512];
    const int tid = threadIdx.x;
    if (tid < 256) f[tid] = feats[tid];
    __syncthreads();
    for (int n = 0; n < 2; ++n) {
        float acc = b1[tid];
        for (int k = 0; k < 128; ++k) acc += f[n * 128 + k] * w1[tid * 128 + k];
        h[n * 512 + tid] = fmaxf(acc, 0.f);
    }
    __syncthreads();
    if (tid < 18) {
        const float Ac[9] = {0.5f, 0.5f, 0.002f, 0.01f, 0.01f, 0.05f, 1.0f, 3.0f, 1e-5f};
        const float Bc[9] = {1.5f, 1.5f, 0.008f, 0.1f, 0.06f, 1.8f, 1.6f, 6.0f, 0.0015f};
        const int n = tid / 9, kk = tid % 9;
        float acc = b2[kk];
        for (int i = 0; i < 512; ++i) acc += h[n * 512 + i] * w2[kk * 512 + i];
        float s = 1.f / (1.f + expf(-acc));
        out[3 + n * 9 + kk] = s * (Bc[kk] - Ac[kk]) + Ac[kk];
    }
}

// ---------------------------------------------------------------------------
// Serial RK4 hemodynamic ODE on sample-0 params (out[3..11]).
// ---------------------------------------------------------------------------
__global__ void ode_k(float* __restrict__ out)
{
    if (threadIdx.x != 0 || blockIdx.x != 0) return;
    const float* p = out + 3;
    const float Tc = p[0], Rs = p[1], Rm = p[2], Ra = p[3], Rc = p[4],
                Ca = p[5], Cs = p[6], Cr = p[7], Ls = p[8];
    const float tc_den = 0.2f + 0.15f * Tc;

    float y[5] = {140.f, 8.2f, 77.f, 77.f, 0.f};
    const float dt = 2.0f / 999.f;         // linspace(0, 2, 1000) spacing
    const float hh = dt / 10.f;            // SUBSTEPS = 10

    const int i_edv = (int)(Tc * 1000.f / 2.0f);
    const int i_esv = (int)(tc_den * 1000.f / 2.0f);
    float v1 = (i_edv == 0) ? 140.f : 0.f;
    float v2 = (i_esv == 0) ? 140.f : 0.f;

    auto deriv = [&](float t, const float* yy, float* dy) {
        float tn = t / tc_den;
        float p1 = powf(tn * (1.f / 0.7f), 1.9f);
        float p2 = powf(tn * (1.f / 1.17f), 21.9f);
        float e  = (2.0f - 0.06f) * 1.55f * p1 / (p1 + 1.f) / (p2 + 1.f) + 0.06f;
        float Plv = e * yy[0];
        float m1 = fmaxf(yy[1] - Plv, 0.f);
        float m2 = fmaxf(Plv - yy[3], 0.f);
        dy[0] = m1 / Rm - m2 / Ra;
        dy[1] = (yy[2] - yy[1]) / (Rs * Cr) - m1 / (Cr * Rm);
        dy[2] = (yy[1] - yy[2]) / (Rs * Cs) + yy[4] / Cs;
        dy[3] = -yy[4] / Ca + m2 / (Ca * Ra);
        dy[4] = (yy[3] - yy[2] - Rc * yy[4]) / Ls;
    };

    for (int s = 0; s < 999; ++s) {
        float t = s * dt;
        for (int i = 0; i < 10; ++i) {
            float t0 = t + i * hh;
            float k1[5], k2[5], k3[5], k4[5], yt[5];
            deriv(t0, y, k1);
            #pragma unroll
            for (int j = 0; j < 5; ++j) yt[j] = y[j] + 0.5f * hh * k1[j];
            deriv(t0 + 0.5f * hh, yt, k2);
            #pragma unroll
            for (int j = 0; j < 5; ++j) yt[j] = y[j] + 0.5f * hh * k2[j];
            deriv(t0 + 0.5f * hh, yt, k3);
            #pragma unroll
            for (int j = 0; j < 5; ++j) yt[j] = y[j] + hh * k3[j];
            deriv(t0 + hh, yt, k4);
            #pragma unroll
            for (int j = 0; j < 5; ++j)
                y[j] += (hh / 6.f) * (k1[j] + 2.f * k2[j] + 2.f * k3[j] + k4[j]);
        }
        const int idx = s + 1;
        if (idx == i_edv) v1 = y[0];
        if (idx == i_esv) v2 = y[0];
    }
    out[0] = v1 + 10.f;
    out[1] = v2 + 10.f;
    out[2] = (v1 - v2) / v1 * 100.f;
}

// ---------------------------------------------------------------------------
// Host-side pipeline (graph-capture safe: only kernel launches on `stream`)
// ---------------------------------------------------------------------------
extern "C" void kernel_launch(void* const* d_in, const int* in_sizes, int n_in,
                              void* d_out, int out_size, void* d_ws, size_t ws_size,
                              hipStream_t stream)
{
    (void)in_sizes; (void)n_in; (void)out_size; (void)ws_size;

    const float* x = (const float*)d_in[0];
    const float* w[5]; const float* bb[5]; const float* g[5]; const float* be[5];
    for (int i = 0; i < 5; ++i) {
        w[i]  = (const float*)d_in[1 + 4 * i + 0];
        bb[i] = (const float*)d_in[1 + 4 * i + 1];
        g[i]  = (const float*)d_in[1 + 4 * i + 2];
        be[i] = (const float*)d_in[1 + 4 * i + 3];
    }
    const float* fc1_w = (const float*)d_in[21];
    const float* fc1_b = (const float*)d_in[22];
    const float* fc2_w = (const float*)d_in[23];
    const float* fc2_b = (const float*)d_in[24];
    float* out = (float*)d_out;

    char* ws = (char*)d_ws;
    float* yBuf  = (float*)ws;                                   // <=128 MiB
    float* ping  = (float*)(ws + (size_t)134217728);             // <=16 MiB
    float* pong  = (float*)(ws + (size_t)134217728 + 16777216);  // <=4 MiB
    char*  tail  = ws + (size_t)134217728 + 16777216 + 4194304;
    float* stats = (float*)tail;              // 256 floats
    float* scsh  = (float*)(tail + 1024);     // 256 floats
    float* feats = (float*)(tail + 2048);     // 256 floats
    float* wTr   = (float*)(tail + 3072);     // up to 27*64*128 floats (~864 KiB)

    const int   Cins[5]  = {1, 8, 16, 32, 64};
    const int   Couts[5] = {8, 16, 32, 64, 128};
    const int   Dims[5]  = {128, 64, 32, 16, 8};
    const float* ins[5]  = {x, ping, pong, ping, pong};
    float*      pools[5] = {ping, pong, ping, pong, nullptr};

    for (int L = 0; L < 5; ++L) {
        const int Cin = Cins[L], Cout = Couts[L], Dm = Dims[L];
        const int CinP  = (Cin + 3) & ~3;
        const int CoutP = (Cout + 15) & ~15;
        const long Ngemm = 2L * Dm * Dm * Dm;

        const int packTot = 27 * CinP * CoutP;
        pack_weights<<<(packTot + 255) / 256, 256, 0, stream>>>(
            w[L], wTr, Cin, Cout, CinP, CoutP);
        zero_f32<<<1, 256, 0, stream>>>(stats, 2 * Cout);

        dim3 grid((unsigned)(Ngemm / 64), (unsigned)(CoutP / 16));
        switch (L) {
        case 0: conv3d_bn_wmma<1,   8, 7><<<grid, 128, 0, stream>>>(ins[L], wTr, bb[L], yBuf, stats); break;
        case 1: conv3d_bn_wmma<8,  16, 6><<<grid, 128, 0, stream>>>(ins[L], wTr, bb[L], yBuf, stats); break;
        case 2: conv3d_bn_wmma<16, 32, 5><<<grid, 128, 0, stream>>>(ins[L], wTr, bb[L], yBuf, stats); break;
        case 3: conv3d_bn_wmma<32, 64, 4><<<grid, 128, 0, stream>>>(ins[L], wTr, bb[L], yBuf, stats); break;
        case 4: conv3d_bn_wmma<64,128, 3><<<grid, 128, 0, stream>>>(ins[L], wTr, bb[L], yBuf, stats); break;
        }

        bn_finalize<<<1, 256, 0, stream>>>(stats, g[L], be[L], scsh, Cout,
                                           1.0f / (float)Ngemm);

        if (pools[L]) {
            const int Do = Dm / 2;
            const long tot = 2L * Cout * Do * Do * Do;
            bn_relu_pool_k<<<(unsigned)((tot + 255) / 256), 256, 0, stream>>>(
                yBuf, scsh, pools[L], Cout, Do);
        } else {
            bn_relu_avg_k<<<2 * Cout, 512, 0, stream>>>(yBuf, scsh, feats, Cout);
        }
    }

    head_k<<<1, 512, 0, stream>>>(feats, fc1_w, fc1_b, fc2_w, fc2_b, out);
    ode_k<<<1, 32, 0, stream>>>(out);
}